// Conv2dGEMM_62758062129885
// MI455X (gfx1250) — compile-verified
//
#include <hip/hip_runtime.h>
#include <hip/hip_bf16.h>

typedef float v2f __attribute__((ext_vector_type(2)));
typedef float v8f __attribute__((ext_vector_type(8)));

// Problem constants (from reference)
#define C_IN   128
#define C_OUT  256
#define HH     56
#define WW     56
#define HWSZ   (HH * WW)          // 3136
#define KTOT   (C_IN * 9)         // 1152
#define NPIX   (32 * HWSZ)        // 100352

// Tiling
#define BM 128
#define BN 128
#define BK 16
#define LDP 20                    // padded row stride (floats) in LDS
#define NSTEPS (KTOT / BK)        // 72  (step s -> tap = s>>3, icBase = (s&7)*16)

#define WT_BYTES  ((size_t)C_OUT * KTOT * sizeof(float))   // 1.18 MB
#define ZP_BYTES  256

// ---------------------------------------------------------------------------
// CDNA5 async global->LDS copy helpers (ASYNCcnt-tracked, no VGPR staging).
// LDS dest = vdst VGPR + imm offset; global src = vaddr + the same imm offset
// ---------------------------------------------------------------------------
__device__ __forceinline__ unsigned lds_addr(const void* p) {
    return (unsigned)(uintptr_t)p;            // low 32 bits of generic ptr = LDS byte addr
}
__device__ __forceinline__ void async_b128(unsigned lds, const float* g) {
    asm volatile("global_load_async_to_lds_b128 %0, %1, off" :: "v"(lds), "v"(g) : "memory");
}
__device__ __forceinline__ void async_b128_off16(unsigned lds, const float* g) {
    asm volatile("global_load_async_to_lds_b128 %0, %1, off offset:16" :: "v"(lds), "v"(g) : "memory");
}
__device__ __forceinline__ void async_b32(unsigned lds, const float* g) {
    asm volatile("global_load_async_to_lds_b32 %0, %1, off" :: "v"(lds), "v"(g) : "memory");
}
__device__ __forceinline__ void wait_async0() {
    asm volatile("s_wait_asynccnt 0" ::: "memory");
}

// ---------------------------------------------------------------------------
// One-time weight permutation: w[oc][ic*9 + t] -> wt[oc][t*128 + ic],
// plus zero page used as the im2col halo source for async copies.
// ---------------------------------------------------------------------------
__global__ __launch_bounds__(256)
void wtrans_kernel(const float* __restrict__ w, float* __restrict__ wt,
                   float* __restrict__ zpage) {
    const int idx = blockIdx.x * 256 + threadIdx.x;    // [0, 256*1152)
    const int oc  = idx / KTOT;
    const int k   = idx - oc * KTOT;
    const int ic  = k / 9;
    const int t   = k - ic * 9;
    wt[(size_t)oc * KTOT + t * C_IN + ic] = w[idx];
    if (idx < (int)(ZP_BYTES / sizeof(float))) zpage[idx] = 0.0f;
}

// ---------------------------------------------------------------------------
// Implicit-GEMM conv, fp32 WMMA, 128x128 tile, double-buffered LDS.
// USE_WT: permuted weights + async global->LDS staging (CDNA5 path).
// else  : strided weights, register staging (fallback when ws too small).
// ---------------------------------------------------------------------------
template <bool USE_WT>
__global__ __launch_bounds__(256)
void Conv2dGEMM_wmma_f32(const float* __restrict__ x,
                         const float* __restrict__ wsrc,
                         const float* __restrict__ bias,
                         float* __restrict__ out,
                         const float* __restrict__ zpage) {
    __shared__ __align__(16) float As[2][BM * LDP];  // [m][k] weight tile
    __shared__ __align__(16) float Bs[2][BN * LDP];  // [n][k] im2col x tile

    const int tid  = threadIdx.x;
    const int lane = tid & 31;
    const int wave = tid >> 5;
    const int mWave = (wave & 3) * 32;   // 4 M sub-tiles of 32
    const int nWave = (wave >> 2) * 64;  // 2 N sub-tiles of 64

    const int blockM = blockIdx.x * BM;  // grid.x = 2
    const int blockN = blockIdx.y * BN;  // grid.y = 784

    // ---- A-tile mapping: 2048 floats, 8 per thread
    const int mA = tid >> 1;             // 0..127
    const int hA = tid & 1;              // which 8-wide half of the 16-k strip
    const float* wRow = wsrc + (size_t)(blockM + mA) * KTOT;

    // ---- B-tile mapping: 2048 floats, 8 per thread (one pixel, 8 channels)
    const int bn    = tid & 127;         // pixel column
    const int bkg   = tid >> 7;          // 0/1 -> k-local 8-group
    const int gn    = blockN + bn;
    const int batch = gn / HWSZ;
    const int remP  = gn - batch * HWSZ;
    const int oh    = remP / WW;
    const int ow    = remP - oh * WW;
    const float* xb = x + (size_t)batch * (C_IN * HWSZ);

    float aReg[8], bReg[8];              // used only by fallback path

    // issue async copies for step s into LDS buffer `buf`
    auto issueTiles = [&](int s, int buf) {
        const int t   = s >> 3;          // tap 0..8 (scalar)
        const int icb = (s & 7) << 4;    // ic base  (scalar)
        // A: two b128 copies, immediate offset reuses the address registers
        {
            const float*  ga = wRow + s * 16 + hA * 8;
            const unsigned la = lds_addr(&As[buf][mA * LDP + hA * 8]);
            async_b128(la, ga);
            async_b128_off16(la, ga);
        }
        // B: eight b32 copies; halo lanes redirected to the zero page
        {
            const int  kh  = t / 3;
            const int  kw  = t - kh * 3;
            const int  ih  = oh + kh - 1;
            const int  iw  = ow + kw - 1;
            const bool ok  = ((unsigned)ih < (unsigned)HH) && ((unsigned)iw < (unsigned)WW);
            const float* src0  = ok ? (xb + (icb + bkg * 8) * HWSZ + ih * WW + iw) : zpage;
            const int    strd  = ok ? HWSZ : 0;
            const unsigned lb  = lds_addr(&Bs[buf][bn * LDP + bkg * 8]);
#pragma unroll
            for (int j = 0; j < 8; ++j)
                async_b32(lb + j * 4, src0 + j * strd);
        }
    };

    // fallback: synchronous register staging (strided weights)
    auto loadTilesSync = [&](int s) {
        const int t   = s >> 3;
        const int icb = (s & 7) << 4;
#pragma unroll
        for (int j = 0; j < 8; ++j)
            aReg[j] = wRow[(icb + hA * 8 + j) * 9 + t];
        const int  kh  = t / 3;
        const int  kw  = t - kh * 3;
        const int  ih  = oh + kh - 1;
        const int  iw  = ow + kw - 1;
        const bool ok  = ((unsigned)ih < (unsigned)HH) && ((unsigned)iw < (unsigned)WW);
        const float okf = ok ? 1.0f : 0.0f;
        const int  base = ok ? (ih * WW + iw) : 0;
#pragma unroll
        for (int j = 0; j < 8; ++j)
            bReg[j] = xb[(icb + bkg * 8 + j) * HWSZ + base] * okf;
    };
    auto storeTilesSync = [&](int buf) {
        *(float4*)(&As[buf][mA * LDP + hA * 8])     = make_float4(aReg[0], aReg[1], aReg[2], aReg[3]);
        *(float4*)(&As[buf][mA * LDP + hA * 8 + 4]) = make_float4(aReg[4], aReg[5], aReg[6], aReg[7]);
        *(float4*)(&Bs[buf][bn * LDP + bkg * 8])     = make_float4(bReg[0], bReg[1], bReg[2], bReg[3]);
        *(float4*)(&Bs[buf][bn * LDP + bkg * 8 + 4]) = make_float4(bReg[4], bReg[5], bReg[6], bReg[7]);
    };

    v8f acc[2][4];
#pragma unroll
    for (int mt = 0; mt < 2; ++mt)
#pragma unroll
        for (int nt = 0; nt < 4; ++nt) acc[mt][nt] = (v8f){};

    // fragment addressing (f32 WMMA 16x16x4, wave32):
    // lanes 0-15 hold K0/K1, lanes 16-31 hold K2/K3
    const int koff = (lane >> 4) << 1;
    const int lm   = lane & 15;

    if (USE_WT) {
        issueTiles(0, 0);
        wait_async0();
    } else {
        loadTilesSync(0);
        storeTilesSync(0);
    }
    __syncthreads();

    for (int s = 0; s < NSTEPS; ++s) {
        const int cur = s & 1;
        if (USE_WT) {
            if (s + 1 < NSTEPS) issueTiles(s + 1, (s + 1) & 1);  // overlap with WMMA
        } else {
            if (s + 1 < NSTEPS) loadTilesSync(s + 1);
        }

        const float* Ac = &As[cur][0];
        const float* Bc = &Bs[cur][0];
#pragma unroll
        for (int kk = 0; kk < 4; ++kk) {
            const int kb = kk * 4 + koff;
            v2f aF[2], bF[4];
#pragma unroll
            for (int mt = 0; mt < 2; ++mt)
                aF[mt] = *(const v2f*)(Ac + (mWave + mt * 16 + lm) * LDP + kb);
#pragma unroll
            for (int nt = 0; nt < 4; ++nt)
                bF[nt] = *(const v2f*)(Bc + (nWave + nt * 16 + lm) * LDP + kb);
#pragma unroll
            for (int mt = 0; mt < 2; ++mt)
#pragma unroll
                for (int nt = 0; nt < 4; ++nt)
                    acc[mt][nt] = __builtin_amdgcn_wmma_f32_16x16x4_f32(
                        false, aF[mt], false, bF[nt], (short)0, acc[mt][nt], false, false);
        }

        if (USE_WT) {
            wait_async0();                      // next buffer's copies landed
        } else {
            if (s + 1 < NSTEPS) storeTilesSync((s + 1) & 1);
        }
        __syncthreads();
    }

    // ---- epilogue: C layout -> out[n, oc, oh, ow] with fused bias
#pragma unroll
    for (int mt = 0; mt < 2; ++mt) {
        const int ocb = blockM + mWave + mt * 16 + ((lane >> 4) << 3);
#pragma unroll
        for (int nt = 0; nt < 4; ++nt) {
            const int nOut = blockN + nWave + nt * 16 + lm;
            const int bo   = nOut / HWSZ;
            const int ohw  = nOut - bo * HWSZ;
#pragma unroll
            for (int r = 0; r < 8; ++r) {
                const int oc = ocb + r;
                out[(size_t)(bo * C_OUT + oc) * HWSZ + ohw] = acc[mt][nt][r] + bias[oc];
            }
        }
    }
}

extern "C" void kernel_launch(void* const* d_in, const int* in_sizes, int n_in,
                              void* d_out, int out_size, void* d_ws, size_t ws_size,
                              hipStream_t stream) {
    const float* x    = (const float*)d_in[0];
    const float* wgt  = (const float*)d_in[1];
    const float* bias = (const float*)d_in[2];
    float* out        = (float*)d_out;

    dim3 grid(C_OUT / BM, NPIX / BN);   // (2, 784)
    dim3 block(256);

    if (ws_size >= WT_BYTES + ZP_BYTES) {
        float* wt    = (float*)d_ws;
        float* zpage = (float*)((char*)d_ws + WT_BYTES);
        wtrans_kernel<<<dim3((C_OUT * KTOT) / 256), dim3(256), 0, stream>>>(wgt, wt, zpage);
        Conv2dGEMM_wmma_f32<true><<<grid, block, 0, stream>>>(x, wt, bias, out, zpage);
    } else {
        Conv2dGEMM_wmma_f32<false><<<grid, block, 0, stream>>>(x, wgt, bias, out, nullptr);
    }
}